// CollisionLoss_55327768708655
// MI455X (gfx1250) — compile-verified
//
#include <hip/hip_runtime.h>
#include <math.h>

#define F_FRAMES 6
#define N_BOXES  1000000
#define TOTAL_BOXES (F_FRAMES * N_BOXES)
#define BLOCK 256
#define NBLK  2048          // partial count; 2048*4B = 8KB of d_ws
#define HALF_W 1.175f       // (1.85 + 0.5) / 2
#define HALF_H 2.292f       // (4.084 + 0.5) / 2

typedef __attribute__((ext_vector_type(2))) float v2f;
typedef __attribute__((ext_vector_type(8))) float v8f;

// ---------------------------------------------------------------------------
// Kernel 1: streaming AABB-overlap reduction.  Memory-bound: 2x
// global_load_b128 per box (8 contiguous f32), per-lane f32 accumulate,
// deterministic LDS tree reduction, one partial per block.
// ---------------------------------------------------------------------------
__global__ void collision_partial_kernel(const float* __restrict__ traj,      // (1,F,2)
                                         const float* __restrict__ gt,        // (1,F,3)
                                         const float* __restrict__ corners,   // (F,N,4,2)
                                         const unsigned char* __restrict__ mask, // (F,N) bool
                                         float* __restrict__ partials)        // (NBLK)
{
    __shared__ float s_xa1[F_FRAMES], s_xa2[F_FRAMES];
    __shared__ float s_ya1[F_FRAMES], s_ya2[F_FRAMES];
    __shared__ float red[BLOCK];

    const int tid = threadIdx.x;

    // Per-block: compute the 6 ego AABBs analytically (exact AABB of the
    // rotated rectangle: extent = |c|*hw + |s|*hh).
    if (tid < F_FRAMES) {
        const float x  = traj[tid * 2 + 0];
        const float y  = traj[tid * 2 + 1];
        const float th = gt[tid * 3 + 2];
        const float c  = fabsf(cosf(th));
        const float s  = fabsf(sinf(th));
        const float ex = c * HALF_W + s * HALF_H;
        const float ey = s * HALF_W + c * HALF_H;
        s_xa1[tid] = x + ex;
        s_xa2[tid] = x - ex;
        s_ya1[tid] = y + ey;
        s_ya2[tid] = y - ey;
    }
    __syncthreads();

    const float4* __restrict__ c4 = reinterpret_cast<const float4*>(corners);

    float acc = 0.0f;
    for (int i = blockIdx.x * BLOCK + tid; i < TOTAL_BOXES; i += NBLK * BLOCK) {
        // Box i: corners (x0,y0,x1,y1),(x2,y2,x3,y3) -> two b128 loads.
        const float4 p = c4[2 * i + 0];
        const float4 q = c4[2 * i + 1];
        const int f = i / N_BOXES;

        const float xb1 = fmaxf(fmaxf(p.x, p.z), fmaxf(q.x, q.z));
        const float xb2 = fminf(fminf(p.x, p.z), fminf(q.x, q.z));
        const float yb1 = fmaxf(fmaxf(p.y, p.w), fmaxf(q.y, q.w));
        const float yb2 = fminf(fminf(p.y, p.w), fminf(q.y, q.w));

        const float w = fmaxf(fminf(s_xa1[f], xb1) - fmaxf(s_xa2[f], xb2), 0.0f);
        const float h = fmaxf(fminf(s_ya1[f], yb1) - fmaxf(s_ya2[f], yb2), 0.0f);

        acc += mask[i] ? (w * h) : 0.0f;
    }

    // Deterministic block tree reduction.
    red[tid] = acc;
    __syncthreads();
    for (int s = BLOCK / 2; s > 0; s >>= 1) {
        if (tid < s) red[tid] += red[tid + s];
        __syncthreads();
    }
    if (tid == 0) partials[blockIdx.x] = red[0];
}

// ---------------------------------------------------------------------------
// Kernel 2: one wave32 (EXEC all ones, uniform control flow) finishes the
// reduction.  Lane-strided f32 sums -> pack 32 lane partials into the A
// matrix of V_WMMA_F32_16X16X4_F32 (v2f/lane: lanes 0-15 = K=0, lanes
// 16-31 = K=2) and multiply by an all-ones B: D row m = lane[m]+lane[m+16],
// exact fp32 FMA.  Sum the 8 D VGPRs per lane, combine lane 0 + lane 16.
// ---------------------------------------------------------------------------
__global__ void collision_final_wmma(const float* __restrict__ partials,
                                     float* __restrict__ out)
{
    __shared__ float tmp[32];
    const int lane = threadIdx.x;   // 0..31, single wave

    float s = 0.0f;
    for (int i = lane; i < NBLK; i += 32)   // NBLK % 32 == 0: uniform trips
        s += partials[i];

    v2f a; a.x = s;    a.y = 0.0f;
    v2f b; b.x = 1.0f; b.y = 1.0f;          // all-ones B: layout-independent
    v8f c = {};
    v8f d = __builtin_amdgcn_wmma_f32_16x16x4_f32(
        /*neg_a=*/false, a, /*neg_b=*/false, b,
        /*c_mod=*/(short)0, c, /*reuse_a=*/false, /*reuse_b=*/false);

    float t = d[0] + d[1] + d[2] + d[3] + d[4] + d[5] + d[6] + d[7];
    tmp[lane] = t;
    __syncthreads();
    if (lane == 0)
        out[0] = (tmp[0] + tmp[16]) * 1.0f;  // WEIGHT = 1.0
}

// ---------------------------------------------------------------------------
extern "C" void kernel_launch(void* const* d_in, const int* in_sizes, int n_in,
                              void* d_out, int out_size, void* d_ws, size_t ws_size,
                              hipStream_t stream) {
    const float*         traj    = (const float*)d_in[0];          // sdc_traj_all (1,F,2)
    const float*         gt      = (const float*)d_in[1];          // sdc_planning_gt (1,F,3)
    /* d_in[2] = sdc_planning_gt_mask: unused by the reference */
    const float*         corners = (const float*)d_in[3];          // future_gt_corners (F,N,4,2)
    const unsigned char* bmask   = (const unsigned char*)d_in[4];  // box_mask (F,N) bool (1B)

    float* partials = (float*)d_ws;     // NBLK floats
    float* outp     = (float*)d_out;    // 1 float

    collision_partial_kernel<<<NBLK, BLOCK, 0, stream>>>(traj, gt, corners, bmask, partials);
    collision_final_wmma<<<1, 32, 0, stream>>>(partials, outp);
}